// InterClassLoss_53137335386663
// MI455X (gfx1250) — compile-verified
//
#include <hip/hip_runtime.h>
#include <hip/hip_bf16.h>

typedef __attribute__((ext_vector_type(2))) float v2f;
typedef __attribute__((ext_vector_type(8))) float v8f;

#define B_    8
#define C_    256
#define HW_   16384
#define CO_   21
#define NSEG_ 168      // B_*CO_
#define PAD_  176      // 11 tiles of 16
#define EPS_  1e-12f

// ---------------------------------------------------------------------------
// k1: per-pixel pseudo label (argmax over 21 channels, masked) + class counts
// ---------------------------------------------------------------------------
__global__ void k1_pseudo(const float* __restrict__ out_old,
                          const int*   __restrict__ labels,
                          const int*   __restrict__ noc_p,
                          int*         __restrict__ seg_cls,
                          unsigned int* __restrict__ cnt) {
    int idx = blockIdx.x * blockDim.x + threadIdx.x;   // 0 .. B*HW-1
    if (idx >= B_ * HW_) return;
    int b = idx / HW_;
    int p = idx - b * HW_;
    const float* base = out_old + (size_t)b * CO_ * HW_ + p;
    float best = base[0];
    int   bi   = 0;
#pragma unroll
    for (int c = 1; c < CO_; ++c) {
        float v = base[(size_t)c * HW_];
        if (v > best) { best = v; bi = c; }            // strict > keeps first max (argmax semantics)
    }
    int noc = *noc_p;
    int cls = (labels[idx] < noc) ? bi : 0;
    seg_cls[idx] = cls;
    atomicAdd(&cnt[b * CO_ + cls], 1u);
}

// ---------------------------------------------------------------------------
// k2: stream one (b,c) plane per block; per-wave LDS bins; unique-writer store
// ---------------------------------------------------------------------------
__global__ void k2_segsum(const float* __restrict__ feat,
                          const float* __restrict__ feat_old,
                          const int*   __restrict__ seg_cls,
                          float*       __restrict__ sum_cur,
                          float*       __restrict__ sum_ref) {
    __shared__ float s_cur[8][CO_];
    __shared__ float s_ref[8][CO_];
    int bid  = blockIdx.x;          // b*C + c
    int b    = bid / C_;
    int c    = bid - b * C_;
    int t    = threadIdx.x;         // 0..255 (8 waves of 32)
    int wave = t >> 5;

    for (int i = t; i < 8 * CO_; i += 256) {
        ((float*)s_cur)[i] = 0.f;
        ((float*)s_ref)[i] = 0.f;
    }
    __syncthreads();

    const float* f  = feat     + (size_t)bid * HW_;
    const float* fo = feat_old + (size_t)bid * HW_;
    const int*   sc = seg_cls  + b * HW_;
    for (int p = t; p < HW_; p += 256) {
        int cls = sc[p];
        atomicAdd(&s_cur[wave][cls], f[p]);
        atomicAdd(&s_ref[wave][cls], fo[p]);
    }
    __syncthreads();

    // each (row = b*CO+cls, col = c) has exactly one writer block -> plain store
    if (t < CO_) {
        float a = 0.f;
        for (int w = 0; w < 8; ++w) a += s_cur[w][t];
        sum_cur[(b * CO_ + t) * C_ + c] = a;
    } else if (t >= 32 && t < 32 + CO_) {
        int cls = t - 32;
        float a = 0.f;
        for (int w = 0; w < 8; ++w) a += s_ref[w][cls];
        sum_ref[(b * CO_ + cls) * C_ + c] = a;
    }
}

// ---------------------------------------------------------------------------
// k3: mean -> L2 normalize, write padded prototype matrices + validity + K
// ---------------------------------------------------------------------------
__global__ void k3_norm(const float* __restrict__ sum_cur,
                        const float* __restrict__ sum_ref,
                        const unsigned int* __restrict__ cnt,
                        float* __restrict__ p_cur,
                        float* __restrict__ p_ref,
                        float* __restrict__ validf,
                        float* __restrict__ kcount) {
    __shared__ float red[256];
    int row = blockIdx.x;           // 0..PAD_-1
    int c   = threadIdx.x;          // 0..255 (== C_)
    if (row >= NSEG_) {
        p_cur[row * C_ + c] = 0.f;
        p_ref[row * C_ + c] = 0.f;
        if (c == 0) validf[row] = 0.f;
        return;
    }
    float n     = (float)cnt[row];
    float denom = fmaxf(n, 1.f);
    float mc = sum_cur[row * C_ + c] / denom;
    float mr = sum_ref[row * C_ + c] / denom;

    red[c] = mc * mc; __syncthreads();
    for (int s = 128; s > 0; s >>= 1) { if (c < s) red[c] += red[c + s]; __syncthreads(); }
    float nc = sqrtf(red[0]); __syncthreads();

    red[c] = mr * mr; __syncthreads();
    for (int s = 128; s > 0; s >>= 1) { if (c < s) red[c] += red[c + s]; __syncthreads(); }
    float nr = sqrtf(red[0]);

    p_cur[row * C_ + c] = mc / fmaxf(nc, EPS_);
    p_ref[row * C_ + c] = mr / fmaxf(nr, EPS_);

    if (c == 0) {
        bool valid = (n > 0.f) && (row % CO_ != 0);    // drop class 0 like cls_old[1:]
        validf[row] = valid ? 1.f : 0.f;
        if (valid) atomicAdd(kcount, 1.f);
    }
}

// ---------------------------------------------------------------------------
// k4: one wave per 16x16 Gram tile; v_wmma_f32_16x16x4_f32 over K=256,
//     for both P_cur and P_ref; fused masked squared-diff accumulation.
//     A (16x4 f32): lane L holds M=L%16, K = k0 + 2*(L>=16) + {0,1}
//     B (4x16 f32): lane L holds N=L%16, same K pair  (B[k][n] = P[j0+n][k])
//     C/D (16x16 f32): VGPR v, lane L -> M = v + 8*(L/16), N = L%16
// ---------------------------------------------------------------------------
__global__ void k4_gram_wmma(const float* __restrict__ p_cur,
                             const float* __restrict__ p_ref,
                             const float* __restrict__ validf,
                             float* __restrict__ sqacc) {
    int ti = blockIdx.x, tj = blockIdx.y;
    int lane = threadIdx.x;         // 0..31
    int lm   = lane & 15;
    int half = lane >> 4;
    int i0 = ti * 16, j0 = tj * 16;

    v8f acc_c = {};
    v8f acc_r = {};
    const float* Ac = p_cur + (size_t)(i0 + lm) * C_;
    const float* Bc = p_cur + (size_t)(j0 + lm) * C_;
    const float* Ar = p_ref + (size_t)(i0 + lm) * C_;
    const float* Br = p_ref + (size_t)(j0 + lm) * C_;

    for (int k0 = 0; k0 < C_; k0 += 4) {
        int kk = k0 + 2 * half;     // even -> 8-byte aligned vector loads
        v2f a_c = *(const v2f*)(Ac + kk);
        v2f b_c = *(const v2f*)(Bc + kk);
        v2f a_r = *(const v2f*)(Ar + kk);
        v2f b_r = *(const v2f*)(Br + kk);
        acc_c = __builtin_amdgcn_wmma_f32_16x16x4_f32(
            false, a_c, false, b_c, (short)0, acc_c, false, false);
        acc_r = __builtin_amdgcn_wmma_f32_16x16x4_f32(
            false, a_r, false, b_r, (short)0, acc_r, false, false);
    }

    float wj = validf[j0 + lm];
    float local = 0.f;
#pragma unroll
    for (int v = 0; v < 8; ++v) {
        int i = i0 + v + 8 * half;
        float w = validf[i] * wj;   // padded rows have validf == 0
        float d = acc_c[v] - acc_r[v];
        local += w * d * d;
    }
    atomicAdd(sqacc, local);
}

// ---------------------------------------------------------------------------
// k5: finalize scalar loss
// ---------------------------------------------------------------------------
__global__ void k5_final(const float* __restrict__ kcount,
                         const float* __restrict__ sqacc,
                         float* __restrict__ out) {
    float k  = *kcount;
    float sq = *sqacc;
    float kk = fmaxf(k * k, 1.f);
    out[0] = (k > 0.f) ? (sq / kk) : 0.f;   // FACTOR == 1.0
}

// ---------------------------------------------------------------------------
extern "C" void kernel_launch(void* const* d_in, const int* in_sizes, int n_in,
                              void* d_out, int out_size, void* d_ws, size_t ws_size,
                              hipStream_t stream) {
    const float* features     = (const float*)d_in[0];
    const float* features_old = (const float*)d_in[1];
    const float* outputs_old  = (const float*)d_in[2];
    // d_in[3] = prototypes (unused by forward)
    const int*   labels       = (const int*)d_in[4];
    const int*   noc          = (const int*)d_in[5];

    char* ws = (char*)d_ws;
    // workspace layout (bytes)
    int*          seg_cls = (int*)(ws + 0);               // 131072 * 4 = 524288
    unsigned int* cnt     = (unsigned int*)(ws + 524288); // 168 u32 (padded 1024)
    float*        sum_cur = (float*)(ws + 525312);        // 168*256*4 = 172032
    float*        sum_ref = (float*)(ws + 697344);        // 172032
    float*        p_cur   = (float*)(ws + 869376);        // 176*256*4 = 180224
    float*        p_ref   = (float*)(ws + 1049600);       // 180224
    float*        validf  = (float*)(ws + 1229824);       // 176*4 = 704
    float*        kcount  = (float*)(ws + 1230528);       // 4
    float*        sqacc   = (float*)(ws + 1230532);       // 4

    // zero only the accumulators touched by atomics (everything else is fully
    // overwritten each call -> deterministic, graph-capture safe)
    hipMemsetAsync(cnt, 0, 1024, stream);
    hipMemsetAsync(kcount, 0, 8, stream);

    k1_pseudo<<<(B_ * HW_) / 256, 256, 0, stream>>>(outputs_old, labels, noc,
                                                    seg_cls, cnt);
    k2_segsum<<<B_ * C_, 256, 0, stream>>>(features, features_old, seg_cls,
                                           sum_cur, sum_ref);
    k3_norm<<<PAD_, C_, 0, stream>>>(sum_cur, sum_ref, cnt,
                                     p_cur, p_ref, validf, kcount);
    k4_gram_wmma<<<dim3(PAD_ / 16, PAD_ / 16), 32, 0, stream>>>(p_cur, p_ref,
                                                                validf, sqacc);
    k5_final<<<1, 1, 0, stream>>>(kcount, sqacc, (float*)d_out);
}